// Encoder_54898271977784
// MI455X (gfx1250) — compile-verified
//
#include <hip/hip_runtime.h>

typedef __attribute__((ext_vector_type(2))) float v2f;
typedef __attribute__((ext_vector_type(8))) float v8f;

#define WAVES_PER_BLOCK 8
// per-wave LDS: h1 16x52 (832) + h2 16x34 (544) + h3 16x18 (288) floats
#define LDS_PER_WAVE (832 + 544 + 288)

// single v_med3_f32 ReLU (avoids fmaxf's canonicalize+max pair)
__device__ __forceinline__ float relu1(float a) {
    return __builtin_amdgcn_fmed3f(a, 0.0f, __builtin_inff());
}

__global__ __launch_bounds__(256) void encoder_mlp_wmma(
    const float* __restrict__ x,  const float* __restrict__ c,
    const float* __restrict__ W1, const float* __restrict__ b1,
    const float* __restrict__ W2, const float* __restrict__ b2,
    const float* __restrict__ W3, const float* __restrict__ b3,
    const float* __restrict__ Wmu, const float* __restrict__ bmu,
    const float* __restrict__ Wlv, const float* __restrict__ blv,
    float* __restrict__ out, int Bn)
{
    __shared__ float lds[WAVES_PER_BLOCK * LDS_PER_WAVE];

    const int lane = threadIdx.x & 31;
    const int wave = threadIdx.x >> 5;
    const int hi   = lane >> 4;      // 0: lanes 0-15, 1: lanes 16-31
    const int sl   = lane & 15;      // sample-in-tile (A/C rows) or N column (B/C cols)

    float* h1 = &lds[wave * LDS_PER_WAVE];   // [16][52]
    float* h2 = h1 + 16 * 52;                // [16][34]
    float* h3 = h2 + 16 * 34;                // [16][18]

    // ---------------- weight preload (registers, reused across tiles) --------
    float w1r[15], b1r[3];
    #pragma unroll
    for (int i = 0; i < 15; ++i) w1r[i] = W1[i];
    #pragma unroll
    for (int i = 0; i < 3; ++i)  b1r[i] = b1[i];

    // fc2: B-fragments for 2 N-tiles x 13 K-steps. K layout (f32 16x16x4):
    //   .x -> K = 4k + 2*hi ; .y -> K = 4k + 2*hi + 1  (pad K>=51 with 0)
    v2f w2f[2][13];
    #pragma unroll
    for (int t = 0; t < 2; ++t) {
        const int n = sl + 16 * t;                 // output feature 0..31
        #pragma unroll
        for (int k = 0; k < 13; ++k) {
            const int k0 = 4 * k + 2 * hi;
            v2f w;
            w.x = (k0     < 51) ? W2[n * 51 + k0]     : 0.0f;
            w.y = (k0 + 1 < 51) ? W2[n * 51 + k0 + 1] : 0.0f;
            w2f[t][k] = w;
        }
    }
    // fc3: [16 out x 32 in], 8 K-steps
    v2f w3f[8];
    #pragma unroll
    for (int k = 0; k < 8; ++k) {
        const int k0 = 4 * k + 2 * hi;
        v2f w; w.x = W3[sl * 32 + k0]; w.y = W3[sl * 32 + k0 + 1];
        w3f[k] = w;
    }
    // heads fused into one 16x16 tile: N=0..2 -> Wmu, N=3..5 -> Wlv, else 0
    v2f whf[4];
    {
        const float* wrow = (sl < 3) ? (Wmu + sl * 16)
                          : (sl < 6) ? (Wlv + (sl - 3) * 16) : nullptr;
        #pragma unroll
        for (int k = 0; k < 4; ++k) {
            const int k0 = 4 * k + 2 * hi;
            v2f w;
            w.x = wrow ? wrow[k0]     : 0.0f;
            w.y = wrow ? wrow[k0 + 1] : 0.0f;
            whf[k] = w;
        }
    }
    const float b2v0 = b2[sl];
    const float b2v1 = b2[sl + 16];
    const float b3v  = b3[sl];
    const float bhv  = (sl < 3) ? bmu[sl] : (sl < 6) ? blv[sl - 3] : 0.0f;

    const int ntiles     = (Bn + 15) >> 4;
    const int totalWaves = gridDim.x * WAVES_PER_BLOCK;
    const int gwave      = blockIdx.x * WAVES_PER_BLOCK + wave;

    for (int tile = gwave; tile < ntiles; tile += totalWaves) {
        const long tbase = (long)tile * 16;

        // ---------------- phase 1: fc1 (scalar) -> LDS h1[16][52] -----------
        {
            long smp = tbase + sl;
            if (smp >= Bn) smp = Bn - 1;           // clamp (stores are guarded)
            const int j0 = hi ? 9 : 0;
            const int nj = hi ? 8 : 9;
            for (int jj = 0; jj < nj; ++jj) {
                const int j = j0 + jj;
                const float* xp = x + (smp * 17 + j) * 3;
                const float* cp = c + (smp * 17 + j) * 2;
                const float i0 = xp[0], i1 = xp[1], i2 = xp[2];
                const float i3 = cp[0], i4 = cp[1];
                #pragma unroll
                for (int o = 0; o < 3; ++o) {
                    float a = b1r[o];
                    a = fmaf(w1r[o * 5 + 0], i0, a);
                    a = fmaf(w1r[o * 5 + 1], i1, a);
                    a = fmaf(w1r[o * 5 + 2], i2, a);
                    a = fmaf(w1r[o * 5 + 3], i3, a);
                    a = fmaf(w1r[o * 5 + 4], i4, a);
                    h1[sl * 52 + j * 3 + o] = relu1(a);
                }
            }
            if (hi) h1[sl * 52 + 51] = 0.0f;       // K padding
        }
        // same-wave LDS ops execute in order; only need a compiler barrier
        asm volatile("" ::: "memory");

        // ---------------- phase 2: fc2 via WMMA (K=52 in 13 steps) ----------
        v8f acc0, acc1;
        #pragma unroll
        for (int r = 0; r < 8; ++r) { acc0[r] = b2v0; acc1[r] = b2v1; }
        #pragma unroll
        for (int k = 0; k < 13; ++k) {
            const int kb = 4 * k + 2 * hi;
            v2f a = *(const v2f*)&h1[sl * 52 + kb];
            acc0 = __builtin_amdgcn_wmma_f32_16x16x4_f32(
                       false, a, false, w2f[0][k], (short)0, acc0, false, false);
            acc1 = __builtin_amdgcn_wmma_f32_16x16x4_f32(
                       false, a, false, w2f[1][k], (short)0, acc1, false, false);
        }
        // ReLU + transpose via LDS: C/D layout -> h2[M][N], M = r + 8*hi, N = sl(+16)
        #pragma unroll
        for (int r = 0; r < 8; ++r) {
            h2[(r + 8 * hi) * 34 + sl]      = relu1(acc0[r]);
            h2[(r + 8 * hi) * 34 + sl + 16] = relu1(acc1[r]);
        }
        asm volatile("" ::: "memory");

        // ---------------- phase 3: fc3 via WMMA (K=32 in 8 steps) -----------
        v8f acc2;
        #pragma unroll
        for (int r = 0; r < 8; ++r) acc2[r] = b3v;
        #pragma unroll
        for (int k = 0; k < 8; ++k) {
            const int kb = 4 * k + 2 * hi;
            v2f a = *(const v2f*)&h2[sl * 34 + kb];
            acc2 = __builtin_amdgcn_wmma_f32_16x16x4_f32(
                       false, a, false, w3f[k], (short)0, acc2, false, false);
        }
        #pragma unroll
        for (int r = 0; r < 8; ++r)
            h3[(r + 8 * hi) * 18 + sl] = relu1(acc2[r]);
        asm volatile("" ::: "memory");

        // ---------------- phase 4: fused heads via WMMA (K=16, 4 steps) -----
        v8f acch;
        #pragma unroll
        for (int r = 0; r < 8; ++r) acch[r] = bhv;
        #pragma unroll
        for (int k = 0; k < 4; ++k) {
            const int kb = 4 * k + 2 * hi;
            v2f a = *(const v2f*)&h3[sl * 18 + kb];
            acch = __builtin_amdgcn_wmma_f32_16x16x4_f32(
                       false, a, false, whf[k], (short)0, acch, false, false);
        }

        // ---------------- stores: mu at [0, 3B), log_var at [3B, 6B) --------
        if (sl < 6) {
            #pragma unroll
            for (int r = 0; r < 8; ++r) {
                const long smp = tbase + r + 8 * hi;
                if (smp < Bn) {
                    if (sl < 3) out[smp * 3 + sl] = acch[r];
                    else        out[(long)Bn * 3 + smp * 3 + (sl - 3)] = acch[r];
                }
            }
        }
    }
}

extern "C" void kernel_launch(void* const* d_in, const int* in_sizes, int n_in,
                              void* d_out, int out_size, void* d_ws, size_t ws_size,
                              hipStream_t stream) {
    const float* x   = (const float*)d_in[0];
    const float* c   = (const float*)d_in[1];
    const float* W1  = (const float*)d_in[2];
    const float* b1  = (const float*)d_in[3];
    const float* W2  = (const float*)d_in[4];
    const float* b2  = (const float*)d_in[5];
    const float* W3  = (const float*)d_in[6];
    const float* b3  = (const float*)d_in[7];
    const float* Wmu = (const float*)d_in[8];
    const float* bmu = (const float*)d_in[9];
    const float* Wlv = (const float*)d_in[10];
    const float* blv = (const float*)d_in[11];
    float* out = (float*)d_out;

    const int Bn = in_sizes[0] / 51;   // x is [B,17,3]
    if (Bn <= 0) return;

    const int blocks = 1024;           // grid-stride over ceil(B/16) wave-tiles
    encoder_mlp_wmma<<<blocks, 256, 0, stream>>>(
        x, c, W1, b1, W2, b2, W3, b3, Wmu, bmu, Wlv, blv, out, Bn);
}